// HartleyCosineConv2d_67113158967466
// MI455X (gfx1250) — compile-verified
//
#include <hip/hip_runtime.h>
#include <stdint.h>

// Problem constants (fixed by the reference).
#define HH    256      // H
#define ROWF  256      // floats per full row (W)
#define ROWB  1024     // bytes per full row
#define HALFF 128      // floats per half row (one wave's share of W)
#define HALFB 512      // bytes per half row
#define RING  16       // rows in the per-wave LDS ring
#define PF    16       // prefetch distance in rows
#define WAVES 8        // waves per block (each wave = one plane-half)

// Native clang vector type: works with __builtin_nontemporal_store and
// lowers to b128 loads/stores when 16B-aligned.
typedef float v4f __attribute__((ext_vector_type(4)));

__device__ __forceinline__ v4f v_fma(float a, v4f b, v4f c) {
  v4f r;
  r.x = fmaf(a, b.x, c.x); r.y = fmaf(a, b.y, c.y);
  r.z = fmaf(a, b.z, c.z); r.w = fmaf(a, b.w, c.w);
  return r;
}

// CDNA5 async copy: per-lane 16B, global -> LDS, tracked by ASYNCcnt.
// Input is streamed exactly once -> non-temporal hint keeps it out of L2.
// %0 = VGPR with LDS byte address, %1 = VGPR pair with 64-bit global address.
__device__ __forceinline__ void async_load_512(uint32_t lds_off, const void* g) {
  asm volatile("global_load_async_to_lds_b128 %0, %1, off th:TH_LOAD_NT"
               :: "v"(lds_off), "v"(g) : "memory");
}

__global__ __launch_bounds__(256) void hartley_stencil_kernel(
    const float* __restrict__ x, const float* __restrict__ alpha,
    float* __restrict__ y) {
  __shared__ float ring[WAVES * RING * HALFF];  // 64 KB per block

  const int tid   = threadIdx.x;
  const int wave  = tid >> 5;
  const int lane  = tid & 31;
  const int gid   = blockIdx.x * WAVES + wave;  // 0..4095 plane-halves
  const int plane = gid >> 1;                   // 0..2047 = n*C + c
  const int half  = gid & 1;
  const int c     = plane & 255;

  // alpha row c = [a0(center), a1(+-1), a2(+-2), a3(+-3)], 16B-aligned.
  const v4f A = reinterpret_cast<const v4f*>(alpha)[c];

  const size_t planeOff = (size_t)plane * (HH * ROWF);
  const float* xh = x + planeOff + half * HALFF + lane * 4;  // row r at +r*ROWF
  float*       yh = y + planeOff + half * HALFF + lane * 4;

  // Raw LDS byte address of this wave's ring (low 32 bits of generic shared ptr).
  const uint32_t ringWave =
      (uint32_t)(uintptr_t)(&ring[0]) + (uint32_t)(wave * RING * HALFB);
  const uint32_t laneB = (uint32_t)lane * 16u;
  const float* ringW = &ring[wave * RING * HALFF + lane * 4];  // C view for reads

  // ---- Prologue: async-prefetch rows 0..15 into the ring (16 ops in flight).
#pragma unroll
  for (int r = 0; r < PF; ++r) {
    async_load_512(ringWave + (uint32_t)(r * HALFB) + laneB,
                   (const void*)((const char*)xh + (size_t)r * ROWB));
  }
  // 12 outstanding allowed -> oldest 4 (rows 0..3) are resident.
  asm volatile("s_wait_asynccnt 12" ::: "memory");

  // Sliding window win[k] = x[reflect(h-3+k)], initialized for h=0: rows 3,2,1,0,1,2,3.
  v4f win[7];
#pragma unroll
  for (int k = 0; k < 7; ++k) {
    int rr = k - 3; if (rr < 0) rr = -rr;
    win[k] = *reinterpret_cast<const v4f*>(ringW + rr * HALFF);
  }

  int h = 0;
  // ---- Main loop: h in [0, 240). Prefetch row h+16 (in flight during the
  // FMA body), consume row h+4 after the wait.
#pragma unroll 4
  for (; h < HH - PF; ++h) {
    const int rp = h + PF;  // <= 255
    async_load_512(ringWave + (uint32_t)((rp & (RING - 1)) * HALFB) + laneB,
                   (const void*)((const char*)xh + (size_t)rp * ROWB));

    v4f s = A.x * win[3];
    s = v_fma(A.y, win[2] + win[4], s);
    s = v_fma(A.z, win[1] + win[5], s);
    s = v_fma(A.w, win[0] + win[6], s);
    // Output is written exactly once: stream it past the caches.
    __builtin_nontemporal_store(s, reinterpret_cast<v4f*>(yh + (size_t)h * ROWF));

    // 12 rows (h+5..h+16) may be in flight -> row h+4 is resident.
    asm volatile("s_wait_asynccnt 12" ::: "memory");

#pragma unroll
    for (int k = 0; k < 6; ++k) win[k] = win[k + 1];
    const int rn = h + 4;  // <= 243, no reflection needed here
    win[6] = *reinterpret_cast<const v4f*>(ringW + (rn & (RING - 1)) * HALFF);
  }

  // ---- Tail: everything is resident; reflect at the bottom edge.
  asm volatile("s_wait_asynccnt 0" ::: "memory");
  for (; h < HH; ++h) {
    v4f s = A.x * win[3];
    s = v_fma(A.y, win[2] + win[4], s);
    s = v_fma(A.z, win[1] + win[5], s);
    s = v_fma(A.w, win[0] + win[6], s);
    __builtin_nontemporal_store(s, reinterpret_cast<v4f*>(yh + (size_t)h * ROWF));

    if (h + 1 < HH) {
#pragma unroll
      for (int k = 0; k < 6; ++k) win[k] = win[k + 1];
      int rn = h + 4;
      if (rn >= HH) rn = 2 * HH - 2 - rn;  // reflect; slots for rows >=240 never rewritten
      win[6] = *reinterpret_cast<const v4f*>(ringW + (rn & (RING - 1)) * HALFF);
    }
  }
}

extern "C" void kernel_launch(void* const* d_in, const int* in_sizes, int n_in,
                              void* d_out, int out_size, void* d_ws, size_t ws_size,
                              hipStream_t stream) {
  const float* x     = (const float*)d_in[0];   // (8,256,256,256) f32
  const float* alpha = (const float*)d_in[1];   // (256,4) f32
  float* y           = (float*)d_out;

  // 2048 planes * 2 halves / 8 waves-per-block = 512 blocks of 256 threads.
  dim3 grid(512), block(256);
  hipLaunchKernelGGL(hartley_stencil_kernel, grid, block, 0, stream, x, alpha, y);
}